// DualScalePatchTST_backbone_11682311045242
// MI455X (gfx1250) — compile-verified
//
#include <hip/hip_runtime.h>
#include <math.h>

typedef float v2f __attribute__((ext_vector_type(2)));
typedef float v8f __attribute__((ext_vector_type(8)));

#define BS_   2
#define CIN_  8
#define CTX_  1024
#define TGT_  96
#define DM_   256
#define NH_   16
#define DK_   16
#define FF_   1024
#define NL_   4
#define NF_   256
#define NC_   64
#define STOT_ 320                 // TOTAL tokens
#define BTOT_ (BS_*CIN_)          // 16 sequences
#define MROWS (BTOT_*STOT_)       // 5120 token rows
#define EPS_  1e-5f

// ---------------- wave helpers (wave32) ----------------
__device__ inline float wave_sum(float v) {
#pragma unroll
  for (int o = 16; o > 0; o >>= 1) v += __shfl_xor(v, o, 32);
  return v;
}
__device__ inline float wave_max(float v) {
#pragma unroll
  for (int o = 16; o > 0; o >>= 1) v = fmaxf(v, __shfl_xor(v, o, 32));
  return v;
}

// ---------------- RevIN statistics ----------------
__global__ void revin_stats_kernel(const float* __restrict__ z,
                                   float* __restrict__ meanp, float* __restrict__ stdp) {
  __shared__ float r1[8], r2[8];
  const int bc = blockIdx.x;           // 0..15
  const int t  = threadIdx.x;          // 256 threads
  const float* zr = z + (size_t)bc * CTX_;
  float s = 0.f, q = 0.f;
  for (int i = t; i < CTX_; i += 256) { float x = zr[i]; s += x; q += x * x; }
  s = wave_sum(s); q = wave_sum(q);
  if ((t & 31) == 0) { r1[t >> 5] = s; r2[t >> 5] = q; }
  __syncthreads();
  if (t == 0) {
    float S = 0.f, Q = 0.f;
#pragma unroll
    for (int i = 0; i < 8; ++i) { S += r1[i]; Q += r2[i]; }
    float m = S / (float)CTX_;
    float var = Q / (float)CTX_ - m * m;
    meanp[bc] = m;
    stdp[bc]  = sqrtf(var + EPS_);
  }
}

// ---------------- dual-scale patch embedding + pos ----------------
__global__ void embed_kernel(const float* __restrict__ z,
                             const float* __restrict__ meanp, const float* __restrict__ stdp,
                             const float* __restrict__ rw, const float* __restrict__ rb,
                             const float* __restrict__ Wf, const float* __restrict__ bf,
                             const float* __restrict__ Wc, const float* __restrict__ bcoarse,
                             const float* __restrict__ Wpos, float* __restrict__ U) {
  const size_t idx = (size_t)blockIdx.x * blockDim.x + threadIdx.x; // over 5120*256
  const int d = (int)(idx & (DM_ - 1));
  const size_t rowidx = idx >> 8;          // bc*320 + s
  const int bc = (int)(rowidx / STOT_);
  const int si = (int)(rowidx - (size_t)bc * STOT_);
  const int c  = bc & (CIN_ - 1);
  const float m = meanp[bc], sd = stdp[bc], w = rw[c], bb = rb[c];
  const float* zr = z + (size_t)bc * CTX_;
  float acc;
  if (si < NF_) {                           // fine patches: len 8, stride 4
    acc = bf[d];
    const int t0 = si * 4;
#pragma unroll
    for (int p = 0; p < 8; ++p) {
      int t = t0 + p; if (t > CTX_ - 1) t = CTX_ - 1;   // replication pad
      float zn = (zr[t] - m) / sd * w + bb;
      acc += zn * Wf[p * DM_ + d];
    }
  } else {                                  // coarse patches: len 32, stride 16
    acc = bcoarse[d];
    const int t0 = (si - NF_) * 16;
#pragma unroll 4
    for (int p = 0; p < 32; ++p) {
      int t = t0 + p; if (t > CTX_ - 1) t = CTX_ - 1;
      float zn = (zr[t] - m) / sd * w + bb;
      acc += zn * Wc[p * DM_ + d];
    }
  }
  U[idx] = acc + Wpos[(size_t)si * DM_ + d];
}

// ---------------- generic fp32 WMMA GEMM: C = A(MxK)*B(KxN) + bias ----------------
// EPI: 0 = none, 1 = exact GELU
template <int EPI>
__global__ void gemm_f32_wmma(const float* __restrict__ A, const float* __restrict__ B,
                              const float* __restrict__ bias, float* __restrict__ C,
                              int M, int N, int K) {
  const int wave = threadIdx.x >> 5;         // 8 waves: 2 (M) x 4 (N)
  const int lane = threadIdx.x & 31;
  const int tileM = blockIdx.x * 32 + (wave >> 2) * 16;
  const int tileN = blockIdx.y * 64 + (wave & 3) * 16;
  if (tileM >= M || tileN >= N) return;      // wave-uniform
  const int half = lane >> 4;                // lane group 0/1
  const int lr   = lane & 15;
  const int koff = half * 2;                 // K offset of this lane group
  const float* Arow = A + (size_t)(tileM + lr) * K;    // A frag row = lr (both halves)
  const float* Bcol = B + (tileN + lr);                // B frag col = lr (both halves)
  v8f acc = {};
  for (int k0 = 0; k0 < K; k0 += 4) {
    v2f a = *(const v2f*)(Arow + k0 + koff);           // A[row][k0+koff .. +1]
    v2f b;
    b.x = Bcol[(size_t)(k0 + koff) * N];               // B[k0+koff  ][col]
    b.y = Bcol[(size_t)(k0 + koff + 1) * N];           // B[k0+koff+1][col]
    acc = __builtin_amdgcn_wmma_f32_16x16x4_f32(false, a, false, b,
                                                (short)0, acc, false, false);
  }
  const float bv = bias[tileN + lr];
#pragma unroll
  for (int r = 0; r < 8; ++r) {
    const int row = tileM + r + 8 * half;              // C/D layout: M = r + 8*group
    float x = acc[r] + bv;
    if (EPI == 1) x = 0.5f * x * (1.f + erff(x * 0.70710678118654752f));
    C[(size_t)row * N + tileN + lr] = x;
  }
}

// ---------------- residual scores: S += scale * Q Kt (per b,h) ----------------
__global__ void scores_wmma(const float* __restrict__ Q, const float* __restrict__ Km,
                            float* __restrict__ S, float scale) {
  const int wave = threadIdx.x >> 5;
  const int lane = threadIdx.x & 31;
  const int kTile = blockIdx.y * 8 + wave;             // 0..19 (wave-uniform guard)
  if (kTile >= STOT_ / 16) return;
  const int qTile = blockIdx.x;                        // 0..19
  const int bh = blockIdx.z;                           // 0..255
  const int b = bh >> 4, h = bh & 15;
  const int half = lane >> 4, lr = lane & 15, koff = half * 2;
  const float* qrow = Q + ((size_t)(b * STOT_ + qTile * 16 + lr)) * DM_ + h * DK_;
  const float* krow = Km + ((size_t)(b * STOT_ + kTile * 16 + lr)) * DM_ + h * DK_;
  v8f acc = {};
#pragma unroll
  for (int k0 = 0; k0 < DK_; k0 += 4) {
    v2f a = *(const v2f*)(qrow + k0 + koff);           // A[q-row][k]
    v2f bfrag = *(const v2f*)(krow + k0 + koff);       // B[k][k-col] == K[k-col][k]
    acc = __builtin_amdgcn_wmma_f32_16x16x4_f32(false, a, false, bfrag,
                                                (short)0, acc, false, false);
  }
#pragma unroll
  for (int r = 0; r < 8; ++r) {
    const int qi = qTile * 16 + r + 8 * half;
    const int ki = kTile * 16 + lr;
    float* sp = S + ((size_t)bh * STOT_ + qi) * STOT_ + ki;
    *sp += scale * acc[r];                             // prev-score residual accumulation
  }
}

// ---------------- softmax(row) then O = W * V, one wave per (bh, qi) row ----------------
__global__ void attn_softmax_av(const float* __restrict__ S, const float* __restrict__ V,
                                float* __restrict__ O) {
  __shared__ float wbuf[8 * STOT_];
  const int wave = threadIdx.x >> 5;
  const int lane = threadIdx.x & 31;
  const size_t row = (size_t)blockIdx.x * 8 + wave;    // 0..81919 == bh*320 + qi
  const float* srow = S + row * STOT_;
  float vals[STOT_ / 32];
  float mx = -3.4e38f;
#pragma unroll
  for (int i = 0; i < STOT_ / 32; ++i) { vals[i] = srow[lane + i * 32]; mx = fmaxf(mx, vals[i]); }
  mx = wave_max(mx);
  float sum = 0.f;
  float* wrow = wbuf + wave * STOT_;
#pragma unroll
  for (int i = 0; i < STOT_ / 32; ++i) {
    float e = expf(vals[i] - mx);
    sum += e;
    wrow[lane + i * 32] = e;
  }
  sum = wave_sum(sum);
  __syncthreads();                                     // make wrow visible across lanes
  const int bh = (int)(row / STOT_);
  const int qi = (int)(row - (size_t)bh * STOT_);
  const int b = bh >> 4, h = bh & 15;
  const int d = lane & 15, kh = lane >> 4;             // split K over lane halves
  const float* vbase = V + ((size_t)(b * STOT_ + kh * (STOT_ / 2))) * DM_ + h * DK_ + d;
  float acc = 0.f;
  for (int k = 0; k < STOT_ / 2; ++k)
    acc += wrow[kh * (STOT_ / 2) + k] * vbase[(size_t)k * DM_];
  acc += __shfl_xor(acc, 16, 32);
  if (lane < 16)
    O[((size_t)(b * STOT_ + qi)) * DM_ + h * DK_ + d] = acc / sum;
}

// ---------------- fused residual add + post-LayerNorm (row length 256) ----------------
__global__ void add_layernorm(float* __restrict__ U, const float* __restrict__ X,
                              const float* __restrict__ gs, const float* __restrict__ gb) {
  __shared__ float red[8];
  const int t = threadIdx.x;
  const size_t row = blockIdx.x;
  const size_t idx = row * DM_ + t;
  float val = U[idx] + X[idx];
  float s = wave_sum(val);
  if ((t & 31) == 0) red[t >> 5] = s;
  __syncthreads();
  float m = 0.f;
#pragma unroll
  for (int i = 0; i < 8; ++i) m += red[i];
  m *= (1.f / DM_);
  __syncthreads();
  const float dv = val - m;
  float q = wave_sum(dv * dv);
  if ((t & 31) == 0) red[t >> 5] = q;
  __syncthreads();
  float var = 0.f;
#pragma unroll
  for (int i = 0; i < 8; ++i) var += red[i];
  var *= (1.f / DM_);
  U[idx] = dv * rsqrtf(var + EPS_) * gs[t] + gb[t];
}

// ---------------- head: permuted flatten -> linear -> RevIN denorm ----------------
__global__ void head_kernel(const float* __restrict__ U, const float* __restrict__ Wh,
                            const float* __restrict__ bh,
                            const float* __restrict__ meanp, const float* __restrict__ stdp,
                            const float* __restrict__ rw, const float* __restrict__ rb,
                            float* __restrict__ out) {
  __shared__ float red[8];
  const int oidx = blockIdx.x;                 // 0..1535
  const int bc = oidx / TGT_, t = oidx - bc * TGT_;
  float acc = 0.f;
  for (int kk = threadIdx.x; kk < DM_ * STOT_; kk += 256) {
    const int d = kk / STOT_, s2 = kk - d * STOT_;   // zz[bc][d*320+s] = U[bc*320+s][d]
    acc += U[((size_t)(bc * STOT_ + s2)) * DM_ + d] * Wh[(size_t)kk * TGT_ + t];
  }
  acc = wave_sum(acc);
  if ((threadIdx.x & 31) == 0) red[threadIdx.x >> 5] = acc;
  __syncthreads();
  if (threadIdx.x == 0) {
    float S = 0.f;
#pragma unroll
    for (int i = 0; i < 8; ++i) S += red[i];
    S += bh[t];
    const int c = bc & (CIN_ - 1);
    out[(size_t)bc * TGT_ + t] =
        (S - rb[c]) / (rw[c] + EPS_ * EPS_) * stdp[bc] + meanp[bc];
  }
}

// ---------------- host orchestration ----------------
extern "C" void kernel_launch(void* const* d_in, const int* in_sizes, int n_in,
                              void* d_out, int out_size, void* d_ws, size_t ws_size,
                              hipStream_t stream) {
  const float* z   = (const float*)d_in[0];
  const float* rw  = (const float*)d_in[1];
  const float* rb  = (const float*)d_in[2];
  const float* Wf  = (const float*)d_in[3];
  const float* bf  = (const float*)d_in[4];
  const float* Wc  = (const float*)d_in[5];
  const float* bcz = (const float*)d_in[6];
  const float* Wpos= (const float*)d_in[7];
  const float* WQ  = (const float*)d_in[8];
  const float* bQ  = (const float*)d_in[9];
  const float* WK  = (const float*)d_in[10];
  const float* bK  = (const float*)d_in[11];
  const float* WV  = (const float*)d_in[12];
  const float* bV  = (const float*)d_in[13];
  const float* WO  = (const float*)d_in[14];
  const float* bO  = (const float*)d_in[15];
  const float* l1s = (const float*)d_in[16];
  const float* l1b = (const float*)d_in[17];
  const float* l2s = (const float*)d_in[18];
  const float* l2b = (const float*)d_in[19];
  const float* F1  = (const float*)d_in[20];
  const float* c1  = (const float*)d_in[21];
  const float* F2  = (const float*)d_in[22];
  const float* c2  = (const float*)d_in[23];
  const float* Wh  = (const float*)d_in[24];
  const float* bhp = (const float*)d_in[25];
  float* out = (float*)d_out;

  // workspace layout (floats)
  const size_t MD = (size_t)MROWS * DM_;            // 1,310,720
  float* ws    = (float*)d_ws;
  float* meanp = ws;                                // 16
  float* stdp  = ws + 16;                           // 16
  float* u     = ws + 32;
  float* qb    = u  + MD;
  float* kb    = qb + MD;
  float* vb    = kb + MD;
  float* ob    = vb + MD;
  float* t1    = ob + MD;                           // 5120 x 1024 (FFN hidden / O-proj tmp)
  float* sc    = t1 + (size_t)MROWS * FF_;          // 256 x 320 x 320 residual scores

  // residual scores start at zero each call
  hipMemsetAsync(sc, 0, (size_t)BTOT_ * NH_ * STOT_ * STOT_ * sizeof(float), stream);

  revin_stats_kernel<<<BTOT_, 256, 0, stream>>>(z, meanp, stdp);
  embed_kernel<<<(int)(MD / 256), 256, 0, stream>>>(z, meanp, stdp, rw, rb,
                                                    Wf, bf, Wc, bcz, Wpos, u);

  const dim3 gProj(MROWS / 32, DM_ / 64);           // 160 x 4
  const dim3 gFF1 (MROWS / 32, FF_ / 64);           // 160 x 16
  const dim3 gScore(STOT_ / 16, 3, BTOT_ * NH_);    // 20 x 3 x 256 (8 k-tiles per block)

  for (int l = 0; l < NL_; ++l) {
    const float* WQl = WQ + (size_t)l * DM_ * DM_; const float* bQl = bQ + l * DM_;
    const float* WKl = WK + (size_t)l * DM_ * DM_; const float* bKl = bK + l * DM_;
    const float* WVl = WV + (size_t)l * DM_ * DM_; const float* bVl = bV + l * DM_;
    const float* WOl = WO + (size_t)l * DM_ * DM_; const float* bOl = bO + l * DM_;
    const float* F1l = F1 + (size_t)l * DM_ * FF_; const float* c1l = c1 + l * FF_;
    const float* F2l = F2 + (size_t)l * FF_ * DM_; const float* c2l = c2 + l * DM_;

    gemm_f32_wmma<0><<<gProj, 256, 0, stream>>>(u, WQl, bQl, qb, MROWS, DM_, DM_);
    gemm_f32_wmma<0><<<gProj, 256, 0, stream>>>(u, WKl, bKl, kb, MROWS, DM_, DM_);
    gemm_f32_wmma<0><<<gProj, 256, 0, stream>>>(u, WVl, bVl, vb, MROWS, DM_, DM_);

    scores_wmma<<<gScore, 256, 0, stream>>>(qb, kb, sc, 0.25f);   // SCALE = dk^-0.5
    attn_softmax_av<<<(BTOT_ * NH_ * STOT_) / 8, 256, 0, stream>>>(sc, vb, ob);

    gemm_f32_wmma<0><<<gProj, 256, 0, stream>>>(ob, WOl, bOl, t1, MROWS, DM_, DM_);
    add_layernorm<<<MROWS, 256, 0, stream>>>(u, t1, l1s + l * DM_, l1b + l * DM_);

    gemm_f32_wmma<1><<<gFF1, 256, 0, stream>>>(u, F1l, c1l, t1, MROWS, FF_, DM_);
    gemm_f32_wmma<0><<<gProj, 256, 0, stream>>>(t1, F2l, c2l, ob, MROWS, DM_, FF_);
    add_layernorm<<<MROWS, 256, 0, stream>>>(u, ob, l2s + l * DM_, l2b + l * DM_);
  }

  head_kernel<<<BTOT_ * TGT_, 256, 0, stream>>>(u, Wh, bhp, meanp, stdp, rw, rb, out);
}